// PtModule_76166950027978
// MI455X (gfx1250) — compile-verified
//
#include <hip/hip_runtime.h>

// Native 4-wide float vector -> global_load_b128 / global_store_b128.
typedef float v4f __attribute__((ext_vector_type(4)));

// Exact reference rounding sequence, per lane:
//   r = (x + 2) * 3; r = (r - 5) / 4; out = r * r
// __f*_rn intrinsics forbid fp-contraction so we match JAX's mul-then-sub rounding.
__device__ __forceinline__ float ew1(float x) {
    float r = __fadd_rn(x, 2.0f);
    r = __fmul_rn(r, 3.0f);
    r = __fsub_rn(r, 5.0f);
    r = __fmul_rn(r, 0.25f);   // /4 is exact as *0.25
    return __fmul_rn(r, r);
}

__device__ __forceinline__ v4f ew4(v4f x) {
    v4f o;
    o.x = ew1(x.x);
    o.y = ew1(x.y);
    o.z = ew1(x.z);
    o.w = ew1(x.w);
    return o;
}

__global__ __launch_bounds__(256) void ew_affine_sq_kernel(
    const v4f* __restrict__ in, v4f* __restrict__ out, int n4) {
    int i      = blockIdx.x * blockDim.x + threadIdx.x;
    int stride = gridDim.x * blockDim.x;

    // 4-way unrolled grid-stride: 4 independent non-temporal B128 loads in
    // flight per wave before the dependent compute/store chain.
    for (; i + 3 * stride < n4; i += 4 * stride) {
        v4f a0 = __builtin_nontemporal_load(&in[i]);
        v4f a1 = __builtin_nontemporal_load(&in[i + stride]);
        v4f a2 = __builtin_nontemporal_load(&in[i + 2 * stride]);
        v4f a3 = __builtin_nontemporal_load(&in[i + 3 * stride]);

        v4f r0 = ew4(a0);
        v4f r1 = ew4(a1);
        v4f r2 = ew4(a2);
        v4f r3 = ew4(a3);

        __builtin_nontemporal_store(r0, &out[i]);
        __builtin_nontemporal_store(r1, &out[i + stride]);
        __builtin_nontemporal_store(r2, &out[i + 2 * stride]);
        __builtin_nontemporal_store(r3, &out[i + 3 * stride]);
    }
    // Tail (dead for 8192*8192, kept for safety).
    for (; i < n4; i += stride) {
        v4f a = __builtin_nontemporal_load(&in[i]);
        __builtin_nontemporal_store(ew4(a), &out[i]);
    }
}

extern "C" void kernel_launch(void* const* d_in, const int* in_sizes, int n_in,
                              void* d_out, int out_size, void* d_ws, size_t ws_size,
                              hipStream_t stream) {
    (void)n_in; (void)d_ws; (void)ws_size; (void)out_size;
    const v4f* in = (const v4f*)d_in[0];
    v4f* out      = (v4f*)d_out;

    const int n  = in_sizes[0];   // 8192*8192 = 67,108,864 (divisible by 4)
    const int n4 = n >> 2;        // 16,777,216 float4 elements

    const int block  = 256;                          // 8 wave32s
    int grid = (n4 + block - 1) / block;
    if (grid > 4096) grid = 4096;                    // ~1M threads, 16 v4f each

    ew_affine_sq_kernel<<<grid, block, 0, stream>>>(in, out, n4);
}